// UpsampleBlock_66451734004054
// MI455X (gfx1250) — compile-verified
//
#include <hip/hip_runtime.h>
#include <stdint.h>

typedef __attribute__((ext_vector_type(2))) float v2f;
typedef __attribute__((ext_vector_type(4))) float v4f;
typedef __attribute__((ext_vector_type(8))) float v8f;
typedef int v2i_gcc __attribute__((vector_size(8)));   // matches builtin proto

#define N_Q 32768
#define N_T 8192
#define F_DIM 128
#define TILES (N_T / 16)              // 512 target tiles of 16
#define CHUNK_TILES 128               // tiles staged per LDS chunk (32 KB)
#define CHUNKS (TILES / CHUNK_TILES)  // 4
#define CHUNK_ELEMS (CHUNK_TILES * 32)
#define PRE_PER_THREAD (CHUNK_ELEMS / 256)  // 16
#define PIPE 4                        // tile-loop software pipeline depth
#define BIG_TSQ 3.0e37f

#define AS1 __attribute__((address_space(1)))
#define AS3 __attribute__((address_space(3)))

#if __has_builtin(__builtin_amdgcn_global_load_async_to_lds_b64) && \
    __has_builtin(__builtin_amdgcn_s_wait_asynccnt)
#define HAVE_ASYNC_LDS 1
#else
#define HAVE_ASYNC_LDS 0
#endif

// ---------------------------------------------------------------------------
// Kernel 0: pre-swizzle targets into the V_WMMA_F32_16X16X4_F32 B-operand
// layout.  B is 4x16 (KxN): lanes 0-15 hold rows K0,K1 (tx,ty), lanes 16-31
// hold rows K2,K3 (tz, masked |t|^2).  One float2 per (tile, lane).
// ---------------------------------------------------------------------------
__global__ void pack_targets_kernel(const float* __restrict__ tp,
                                    const uint8_t* __restrict__ tmask,
                                    v2f* __restrict__ Bg) {
    int id = blockIdx.x * blockDim.x + threadIdx.x;   // 0 .. TILES*32-1
    if (id >= TILES * 32) return;
    int tile = id >> 5;
    int lane = id & 31;
    int half = lane >> 4;
    int l16  = lane & 15;
    int n = tile * 16 + l16;                          // target index (column N)
    float x = tp[n * 3 + 0];
    float y = tp[n * 3 + 1];
    float z = tp[n * 3 + 2];
    v2f v;
    if (half == 0) {
        v.x = x;                                      // K0 row
        v.y = y;                                      // K1 row
    } else {
        float tsq = x * x + y * y + z * z;
        v.x = z;                                      // K2 row
        v.y = tmask[n] ? tsq : BIG_TSQ;               // K3 row (mask folded in)
    }
    Bg[id] = v;
}

// ---------------------------------------------------------------------------
// Kernel 1: per-wave 16-query argmin sweep over all 8192 targets using
// V_WMMA_F32_16X16X4_F32:  D = A x B + C  ==  |q|^2 + |t|^2_masked - 2 q.t
//   A row(M) = (-2qx, -2qy, -2qz, 1)   C[M][*] = |q|^2
// LDS chunks are double-buffered via async global->LDS copies (ASYNCcnt);
// the tile loop is software-pipelined PIPE(=unroll) deep so every ds_load
// is consumed one full unrolled body after it issues.
// ---------------------------------------------------------------------------
__global__ __launch_bounds__(256) void nn_argmin_kernel(
        const float* __restrict__ qp,
        const v2f* __restrict__ Bg,
        int* __restrict__ nn_idx) {
    __shared__ v2f sB[2][CHUNK_ELEMS];                // 2 x 32 KB

    int lane = threadIdx.x & 31;
    int wave = threadIdx.x >> 5;                      // 8 waves / block
    int half = lane >> 4;
    int l16  = lane & 15;
    int qbase = blockIdx.x * 128 + wave * 16;

    // A operand: row M = qbase + l16.  Lanes 0-15 carry K0,K1; 16-31 carry K2,K3.
    int qa = qbase + l16;
    float qx = qp[qa * 3 + 0];
    float qy = qp[qa * 3 + 1];
    float qz = qp[qa * 3 + 2];
    float qsq_self = qx * qx + qy * qy + qz * qz;
    v2f a;
    if (half == 0) { a.x = -2.0f * qx; a.y = -2.0f * qy; }
    else           { a.x = -2.0f * qz; a.y = 1.0f;       }

    // C operand: c[r] = |q|^2 of query M = r + 8*half (C/D VGPR-row layout).
    v8f c;
    #pragma unroll
    for (int r = 0; r < 8; ++r)
        c[r] = __shfl(qsq_self, r + 8 * half, 32);

    float mind[8];
    int   mini[8];
    #pragma unroll
    for (int r = 0; r < 8; ++r) { mind[r] = 3.3e38f; mini[r] = 0; }

    // Stage chunk 0 (plain path; cold start, nothing to overlap with).
    #pragma unroll 4
    for (int i = threadIdx.x; i < CHUNK_ELEMS; i += 256)
        sB[0][i] = Bg[i];
    __syncthreads();

    for (int chunk = 0; chunk < CHUNKS; ++chunk) {
        const v2f* buf = sB[chunk & 1];

#if HAVE_ASYNC_LDS
        // Issue async global->LDS copies for the next chunk; they complete
        // in the background (ASYNCcnt) while this chunk's WMMA loop runs.
        if (chunk + 1 < CHUNKS) {
            const v2f* src = Bg + (size_t)(chunk + 1) * CHUNK_ELEMS;
            v2f* dst = sB[(chunk + 1) & 1];
            #pragma unroll
            for (int k = 0; k < PRE_PER_THREAD; ++k) {
                int i = threadIdx.x + k * 256;
                __builtin_amdgcn_global_load_async_to_lds_b64(
                    (AS1 v2i_gcc*)(uintptr_t)(src + i),
                    (AS3 v2i_gcc*)(uint32_t)(uintptr_t)(dst + i),
                    /*offset=*/0, /*cpol=*/0);
            }
        }
#else
        // Fallback: prefetch next chunk into registers now (loads issue and
        // land during the compute loop), drain to LDS after the loop.
        v2f pre[PRE_PER_THREAD];
        if (chunk + 1 < CHUNKS) {
            const v2f* src = Bg + (size_t)(chunk + 1) * CHUNK_ELEMS;
            #pragma unroll
            for (int k = 0; k < PRE_PER_THREAD; ++k)
                pre[k] = src[threadIdx.x + k * 256];
        }
#endif

        int nb = chunk * CHUNK_TILES * 16 + l16;      // this lane's column N base

        // Fill the PIPE-deep B-operand ring.
        v2f bq[PIPE];
        #pragma unroll
        for (int k = 0; k < PIPE; ++k)
            bq[k] = buf[k * 32 + lane];

        #pragma unroll PIPE
        for (int t = 0; t < CHUNK_TILES; ++t) {
            // Prefetch tile t+PIPE (wrapped; tail prefetches are dead but safe).
            v2f bn = buf[((t + PIPE) & (CHUNK_TILES - 1)) * 32 + lane];
            // 8 args: (neg_a, A, neg_b, B, c_mod, C, reuse_a, reuse_b)
            v8f d = __builtin_amdgcn_wmma_f32_16x16x4_f32(
                false, a, false, bq[0], (short)0, c, false, false);
            int nidx = nb + t * 16;
            #pragma unroll
            for (int r = 0; r < 8; ++r) {
                if (d[r] < mind[r]) { mind[r] = d[r]; mini[r] = nidx; }
            }
            #pragma unroll
            for (int k = 0; k < PIPE - 1; ++k)
                bq[k] = bq[k + 1];
            bq[PIPE - 1] = bn;
        }

#if HAVE_ASYNC_LDS
        __builtin_amdgcn_s_wait_asynccnt(0);
#else
        if (chunk + 1 < CHUNKS) {
            v2f* dst = sB[(chunk + 1) & 1];
            #pragma unroll
            for (int k = 0; k < PRE_PER_THREAD; ++k)
                dst[threadIdx.x + k * 256] = pre[k];
        }
#endif
        __syncthreads();
    }

    // Reduce across the 16 lanes that share each query row (xor of bits 0-3
    // stays inside each half of the wave32).  Ties -> lowest index (argmin).
    #pragma unroll
    for (int off = 1; off < 16; off <<= 1) {
        #pragma unroll
        for (int r = 0; r < 8; ++r) {
            float od = __shfl_xor(mind[r], off, 32);
            int   oi = __shfl_xor(mini[r], off, 32);
            if (od < mind[r] || (od == mind[r] && oi < mini[r])) {
                mind[r] = od;
                mini[r] = oi;
            }
        }
    }

    if (l16 == 0) {
        #pragma unroll
        for (int r = 0; r < 8; ++r)
            nn_idx[qbase + r + 8 * half] = mini[r];
    }
}

// ---------------------------------------------------------------------------
// Kernel 2: feature gather with q_mask (float4-wide), out = feat[idx[q]]
// ---------------------------------------------------------------------------
__global__ void gather_feats_kernel(const float* __restrict__ feat,
                                    const int* __restrict__ nn_idx,
                                    const uint8_t* __restrict__ qmask,
                                    float* __restrict__ outf) {
    int gid = blockIdx.x * blockDim.x + threadIdx.x;  // N_Q * 32 threads
    int q  = gid >> 5;
    int c4 = gid & 31;
    if (q >= N_Q) return;
    int idx = qmask[q] ? nn_idx[q] : 0;
    const v4f* src = (const v4f*)(feat + (size_t)idx * F_DIM);
    v4f*       dst = (v4f*)(outf + (size_t)q * F_DIM);
    dst[c4] = src[c4];
}

// Kernel 3: pass-through copy of query_points into the first output slot.
__global__ void copy_qpts_kernel(const float* __restrict__ qp,
                                 float* __restrict__ out) {
    int gid = blockIdx.x * blockDim.x + threadIdx.x;  // N_Q*3/4 float4s
    if (gid >= (N_Q * 3) / 4) return;
    ((v4f*)out)[gid] = ((const v4f*)qp)[gid];
}

extern "C" void kernel_launch(void* const* d_in, const int* in_sizes, int n_in,
                              void* d_out, int out_size, void* d_ws, size_t ws_size,
                              hipStream_t stream) {
    const float*   qp = (const float*)d_in[0];    // [N_Q,3]
    const float*   tp = (const float*)d_in[1];    // [N_T,3]
    const float*   tf = (const float*)d_in[2];    // [N_T,F_DIM]
    const uint8_t* qm = (const uint8_t*)d_in[3];  // [N_Q] bool
    const uint8_t* tm = (const uint8_t*)d_in[4];  // [N_T] bool
    float* out = (float*)d_out;

    // Workspace: packed B operands (128 KB), then per-query argmin (128 KB).
    v2f* Bg = (v2f*)d_ws;
    int* nn = (int*)((char*)d_ws + (size_t)TILES * 32 * sizeof(v2f));

    pack_targets_kernel<<<(TILES * 32 + 255) / 256, 256, 0, stream>>>(tp, tm, Bg);
    nn_argmin_kernel<<<N_Q / 128, 256, 0, stream>>>(qp, Bg, nn);
    copy_qpts_kernel<<<((N_Q * 3) / 4 + 255) / 256, 256, 0, stream>>>(qp, out);
    gather_feats_kernel<<<(N_Q * 32 + 255) / 256, 256, 0, stream>>>(
        tf, nn, qm, out + (size_t)N_Q * 3);
}